// TrajectoryInference_75118978007317
// MI455X (gfx1250) — compile-verified
//
#include <hip/hip_runtime.h>
#include <math.h>

// Problem constants (match the reference)
#define NROWS   8192
#define DIM     256
#define H1      128
#define H2      64
#define NC      50
#define KPAD    64          // coords padded K (50 -> 64, zero-filled)
#define COLS_PER_BLOCK 512  // columns handled by one wave in pass 1
#define EPS     1e-8f

typedef __attribute__((ext_vector_type(2))) float v2f;
typedef __attribute__((ext_vector_type(8))) float v8f;

// ---------------------------------------------------------------------------
// Kernel 1: MLP (Linear+GELU x2, Linear) -> coords[8192][64] (zero padded),
//           sq[row] = sum(coords^2). One row per block, 128 threads.
// ---------------------------------------------------------------------------
__global__ void ti_mlp_kernel(const float* __restrict__ emb,
                              const float* __restrict__ W1, const float* __restrict__ b1,
                              const float* __restrict__ W2, const float* __restrict__ b2,
                              const float* __restrict__ W3, const float* __restrict__ b3,
                              float* __restrict__ coords, float* __restrict__ sqv) {
  __shared__ float x[DIM];
  __shared__ float h1[H1];
  __shared__ float h2[H2];
  __shared__ float red[H2];

  const int row = blockIdx.x;
  const int t = threadIdx.x;  // 128 threads

  x[t]       = emb[row * DIM + t];
  x[t + 128] = emb[row * DIM + t + 128];
  __syncthreads();

  // layer 1: 256 -> 128, exact GELU
  {
    float acc = b1[t];
    #pragma unroll 8
    for (int k = 0; k < DIM; ++k) acc = fmaf(x[k], W1[k * H1 + t], acc);
    acc = 0.5f * acc * (1.0f + erff(acc * 0.70710678118654752f));
    h1[t] = acc;
  }
  __syncthreads();

  // layer 2: 128 -> 64, exact GELU
  if (t < H2) {
    float acc = b2[t];
    #pragma unroll 8
    for (int k = 0; k < H1; ++k) acc = fmaf(h1[k], W2[k * H2 + t], acc);
    acc = 0.5f * acc * (1.0f + erff(acc * 0.70710678118654752f));
    h2[t] = acc;
  }
  __syncthreads();

  // layer 3: 64 -> 50, write zero-padded coords row of width KPAD
  float c = 0.0f;
  if (t < KPAD) {
    if (t < NC) {
      c = b3[t];
      #pragma unroll 8
      for (int k = 0; k < H2; ++k) c = fmaf(h2[k], W3[k * NC + t], c);
    }
    coords[(size_t)row * KPAD + t] = c;  // pads [50,64) with zeros
    red[t] = c * c;
  }
  __syncthreads();
  if (t == 0) {
    float s = 0.0f;
    #pragma unroll
    for (int k = 0; k < KPAD; ++k) s += red[k];
    sqv[row] = s;
  }
}

// ---------------------------------------------------------------------------
// Kernel 2: softmax(diffusion_weights) tiled to w_full[8192]; zero rowsum.
// Single block of 256 threads.
// ---------------------------------------------------------------------------
__global__ void ti_prep_kernel(const float* __restrict__ dw,
                               float* __restrict__ wfull,
                               float* __restrict__ rowsum) {
  __shared__ float w[NC];
  const int t = threadIdx.x;
  if (t == 0) {
    float mx = dw[0];
    for (int k = 1; k < NC; ++k) mx = fmaxf(mx, dw[k]);
    float s = 0.0f;
    for (int k = 0; k < NC; ++k) { float e = __expf(dw[k] - mx); w[k] = e; s += e; }
    float inv = 1.0f / s;
    for (int k = 0; k < NC; ++k) w[k] *= inv;
  }
  __syncthreads();
  for (int idx = t; idx < NROWS; idx += 256) {
    wfull[idx] = w[idx % NC];
    rowsum[idx] = 0.0f;
  }
}

// ---------------------------------------------------------------------------
// Kernel 3: WMMA pairwise pass. One wave (32 threads) per block computes a
// 16-row x 512-col region: Gram tile via V_WMMA_F32_16X16X4_F32, fused
// epilogue d2 -> exp -> column weight, writes unnormalized values, and
// atomically accumulates per-row sums.
//   grid = (8192/COLS_PER_BLOCK, 8192/16), block = 32.
// ---------------------------------------------------------------------------
__global__ void ti_pair_kernel(const float* __restrict__ coords,
                               const float* __restrict__ sqv,
                               const float* __restrict__ wfull,
                               const float* __restrict__ sigma,
                               float* __restrict__ out,
                               float* __restrict__ rowsum) {
  const int lane = threadIdx.x;         // 0..31, wave32, EXEC all ones
  const int m    = lane & 15;           // row-in-tile / col-in-tile index
  const int ksel = lane >> 4;           // which K-half this lane holds

  const int i     = blockIdx.y * 16;                // row strip base
  const int jbase = blockIdx.x * COLS_PER_BLOCK;    // column chunk base

  float sg = sigma[0];
  sg = fminf(fmaxf(sg, 0.1f), 10.0f);
  const float inv2s2 = 1.0f / (2.0f * sg * sg);

  // Preload all 16 A fragments (K=0..63 in steps of 4) for this row strip.
  // Layout per ISA 7.12.2: lanes 0-15 hold K={0,1}, lanes 16-31 hold K={2,3}.
  v2f a[16];
  const float* arow = coords + (size_t)(i + m) * KPAD + 2 * ksel;
  #pragma unroll
  for (int kk = 0; kk < 16; ++kk) a[kk] = *(const v2f*)(arow + 4 * kk);

  // sq for the 8 output rows this lane's C registers map to (M = v + 8*ksel)
  float sqrow[8];
  #pragma unroll
  for (int v = 0; v < 8; ++v) sqrow[v] = sqv[i + v + 8 * ksel];

  float racc[8] = {0.f, 0.f, 0.f, 0.f, 0.f, 0.f, 0.f, 0.f};

  for (int jt = 0; jt < COLS_PER_BLOCK; jt += 16) {
    const int j = jbase + jt;
    // B fragment (K x 16) of rows j..j+15 has the identical layout as an
    // A fragment of those rows, since B = A_j^T for the Gram matrix.
    const float* brow = coords + (size_t)(j + m) * KPAD + 2 * ksel;

    v8f c = {0.f, 0.f, 0.f, 0.f, 0.f, 0.f, 0.f, 0.f};
    #pragma unroll
    for (int kk = 0; kk < 16; ++kk) {
      v2f b = *(const v2f*)(brow + 4 * kk);
      c = __builtin_amdgcn_wmma_f32_16x16x4_f32(
          /*neg_a=*/false, a[kk], /*neg_b=*/false, b,
          /*c_mod=*/(short)0, c, /*reuse_a=*/false, /*reuse_b=*/false);
    }

    // Epilogue: C tile element (VGPR v, lane) = dot(row i+M, row j+N),
    // M = v + 8*ksel, N = m.
    const float sqj = sqv[j + m];
    const float wj  = wfull[j + m];
    float* obase = out + (size_t)(i + 8 * ksel) * NROWS + j + m;
    #pragma unroll
    for (int v = 0; v < 8; ++v) {
      float d2 = sqrow[v] + sqj - 2.0f * c[v];
      d2 = fmaxf(d2, 0.0f);
      float val = __expf(-d2 * inv2s2) * wj;
      racc[v] += val;
      obase[(size_t)v * NROWS] = val;   // 64B-contiguous per 16-lane half
    }
  }

  // Row-sum reduction across the 16 lanes of each half-wave.
  #pragma unroll
  for (int v = 0; v < 8; ++v) {
    float s = racc[v];
    #pragma unroll
    for (int mk = 1; mk < 16; mk <<= 1) s += __shfl_xor(s, mk, 32);
    if (m == 0) atomicAdd(&rowsum[i + v + 8 * ksel], s);
  }
}

// ---------------------------------------------------------------------------
// Kernel 4: in-place row normalization, float4-vectorized grid stride.
// ---------------------------------------------------------------------------
__global__ void ti_norm_kernel(float* __restrict__ out,
                               const float* __restrict__ rowsum) {
  const size_t total = (size_t)NROWS * NROWS / 4;
  float4* o4 = (float4*)out;
  size_t stride = (size_t)gridDim.x * blockDim.x;
  for (size_t v = (size_t)blockIdx.x * blockDim.x + threadIdx.x; v < total; v += stride) {
    const int row = (int)(v >> 11);  // 8192/4 = 2048 float4 per row
    const float inv = 1.0f / (rowsum[row] + EPS);
    float4 t = o4[v];
    t.x *= inv; t.y *= inv; t.z *= inv; t.w *= inv;
    o4[v] = t;
  }
}

// ---------------------------------------------------------------------------
extern "C" void kernel_launch(void* const* d_in, const int* in_sizes, int n_in,
                              void* d_out, int out_size, void* d_ws, size_t ws_size,
                              hipStream_t stream) {
  const float* emb = (const float*)d_in[0];
  const float* W1  = (const float*)d_in[1];
  const float* b1  = (const float*)d_in[2];
  const float* W2  = (const float*)d_in[3];
  const float* b2  = (const float*)d_in[4];
  const float* W3  = (const float*)d_in[5];
  const float* b3  = (const float*)d_in[6];
  const float* dw  = (const float*)d_in[7];
  const float* sig = (const float*)d_in[8];
  float* out = (float*)d_out;

  // Workspace layout (floats): coords[8192*64] | sq[8192] | wfull[8192] | rowsum[8192]
  float* coords = (float*)d_ws;
  float* sqv    = coords + (size_t)NROWS * KPAD;
  float* wfull  = sqv + NROWS;
  float* rowsum = wfull + NROWS;

  // 1) MLP -> coords (padded) + per-row sum of squares
  ti_mlp_kernel<<<NROWS, 128, 0, stream>>>(emb, W1, b1, W2, b2, W3, b3, coords, sqv);

  // 2) softmax + tiling of diffusion weights; zero row sums
  ti_prep_kernel<<<1, 256, 0, stream>>>(dw, wfull, rowsum);

  // 3) WMMA Gram + fused exp/weight epilogue + row-sum atomics
  dim3 grid(NROWS / COLS_PER_BLOCK, NROWS / 16, 1);
  ti_pair_kernel<<<grid, 32, 0, stream>>>(coords, sqv, wfull, sig, out, rowsum);

  // 4) row normalization in place
  ti_norm_kernel<<<16384, 256, 0, stream>>>(out, rowsum);
}